// Exch_72928544686108
// MI455X (gfx1250) — compile-verified
//
#include <hip/hip_runtime.h>

typedef float v2f __attribute__((ext_vector_type(2)));
typedef float v8f __attribute__((ext_vector_type(8)));

#define NX 256
#define NY 256
#define NZ 128
#define NVOX ((size_t)NX * NY * NZ)          // 8388608
#define DX_CONST 5e-09f
#define GEO_THRESH 0.001f
#define NPARTIAL 4096                         // 128 x 32 blocks

// Stage 1: 6-point stencil, per-voxel energy, deterministic block reduction.
// Block (128,2): threadIdx.x = k (full contiguous row, coalesced),
// threadIdx.y picks one of 2 j-rows. blockIdx.x = j-tile, blockIdx.y strides i.
__global__ __launch_bounds__(256) void exch_stencil(
    const float* __restrict__ spin, const float* __restrict__ Ms,
    const float* __restrict__ A, float* __restrict__ partial)
{
  const int k = threadIdx.x;
  const int j = blockIdx.x * 2 + threadIdx.y;
  float acc = 0.0f;

  for (int i = blockIdx.y; i < NX; i += 32) {
    const size_t idx = ((size_t)i * NY + j) * NZ + k;
    const float x0 = spin[idx];
    const float x1 = spin[NVOX + idx];
    const float x2 = spin[2 * NVOX + idx];
    float f0 = 0.0f, f1 = 0.0f, f2 = 0.0f;

    // Zero-padding => out-of-bounds neighbors contribute exactly 0: skip them.
    // In-bounds: term = (x(nb) - x(p)) * geo(nb), geo = (Ms(nb) > thresh).
#define NB(cond, nidx)                                        \
    do {                                                      \
      if (cond) {                                             \
        const size_t nn = (nidx);                             \
        const float g = (Ms[nn] > GEO_THRESH) ? 1.0f : 0.0f;  \
        f0 = fmaf(spin[nn]            - x0, g, f0);           \
        f1 = fmaf(spin[NVOX + nn]     - x1, g, f1);           \
        f2 = fmaf(spin[2 * NVOX + nn] - x2, g, f2);           \
      }                                                       \
    } while (0)

    NB(k > 0,      idx - 1);
    NB(k < NZ - 1, idx + 1);
    NB(j > 0,      idx - NZ);
    NB(j < NY - 1, idx + NZ);
    NB(i > 0,      idx - (size_t)NY * NZ);
    NB(i < NX - 1, idx + (size_t)NY * NZ);
#undef NB

    const float e = f0 * x0 + f1 * x1 + f2 * x2;
    acc = fmaf(A[idx], e, acc);
  }

  // Deterministic fixed-order tree reduction in LDS.
  __shared__ float red[256];
  const int t = threadIdx.y * 128 + threadIdx.x;
  red[t] = acc;
  __syncthreads();
  for (int s = 128; s > 0; s >>= 1) {
    if (t < s) red[t] += red[t + s];
    __syncthreads();
  }
  if (t == 0) partial[blockIdx.y * gridDim.x + blockIdx.x] = red[0];
}

// Stage 2: single wave32. Each of the 64 (lane, A-VGPR) slots accumulates a
// disjoint strided subset of the partials (uniform trip counts => EXEC stays
// all-ones, as V_WMMA requires). Then D = A x ones(4x16) + 0 yields exact f32
// row-sums of A replicated across lanes:
//   D vgpr v, lanes 0-15 hold rowsum(M=v), lanes 16-31 hold rowsum(M=v+8).
// Per-lane sum of the 8 D VGPRs + (lane0 + lane16) = exact total.
__global__ __launch_bounds__(32) void reduce_wmma(
    const float* __restrict__ partial, float* __restrict__ out, int n)
{
  const int lane = threadIdx.x;
  float a0 = 0.0f, a1 = 0.0f;
  for (int idx = lane; idx < n; idx += 64)      a0 += partial[idx];
  for (int idx = lane + 32; idx < n; idx += 64) a1 += partial[idx];

  v2f a; a.x = a0;   a.y = a1;
  v2f b; b.x = 1.0f; b.y = 1.0f;   // all-ones B: layout-independent
  v8f c = {};
  // 8 args: (neg_a, A, neg_b, B, c_mod, C, reuse_a, reuse_b)
  c = __builtin_amdgcn_wmma_f32_16x16x4_f32(
      false, a, false, b, (short)0, c, false, false);

  float tsum = c[0] + c[1] + c[2] + c[3] + c[4] + c[5] + c[6] + c[7];
  const float total = __shfl(tsum, 0, 32) + __shfl(tsum, 16, 32);

  if (lane == 0) {
    out[0] = total * (-DX_CONST / (float)NVOX);
  }
}

extern "C" void kernel_launch(void* const* d_in, const int* in_sizes, int n_in,
                              void* d_out, int out_size, void* d_ws, size_t ws_size,
                              hipStream_t stream) {
  (void)in_sizes; (void)n_in; (void)out_size; (void)ws_size;
  const float* spin = (const float*)d_in[0];
  const float* Ms   = (const float*)d_in[1];
  const float* A    = (const float*)d_in[2];
  float* partial = (float*)d_ws;   // 4096 floats = 16 KB scratch

  dim3 block(128, 2, 1);
  dim3 grid(128, 32, 1);           // j-tiles x i-stride blocks
  exch_stencil<<<grid, block, 0, stream>>>(spin, Ms, A, partial);
  reduce_wmma<<<1, 32, 0, stream>>>(partial, (float*)d_out, NPARTIAL);
}